// RGCNSA_22179211117211
// MI455X (gfx1250) — compile-verified
//
#include <hip/hip_runtime.h>
#include <hip/hip_bf16.h>
#include <math.h>

// ---------------------------------------------------------------------------
// RGCN single-relation-score kernel set for MI455X (gfx1250, wave32).
//
// Reformulation: msg[e] = dot(x[src[e]], W[etype[e]]) with R=16 relations
//   ==>  S = x @ W^T  (N x 16 GEMM, same FLOPs as the edge loop: N*D*R == E*D)
//   then msg[e] = S[src[e], etype[e]]  (one 4-byte gather per edge).
// GEMM uses V_WMMA_F32_16X16X4_F32 (exact f32; the problem is BW-bound:
// ~115 MB HBM traffic vs 0.41 GFLOP -> ~8us roofline at 23.3 TB/s).
// ---------------------------------------------------------------------------

typedef float v2f __attribute__((ext_vector_type(2)));
typedef float v8f __attribute__((ext_vector_type(8)));

#define DFEAT 256
#define NREL  16

// ---- kernel 1: zero accumulators -------------------------------------------
__global__ void rgcn_zero(float* __restrict__ p, int n) {
    int i = blockIdx.x * blockDim.x + threadIdx.x;
    if (i < n) p[i] = 0.0f;
}

// ---- kernel 2: S[N,16] = x[N,256] @ W[16,256]^T via WMMA f32 16x16x4 -------
// One wave computes a 16(node) x 16(relation) tile. K loop: 256/4 = 64 WMMAs.
//
// A (16x4, MxK) layout per ISA 7.12.2: lanes 0-15 hold M=lane (K=0,1 in the
// two VGPRs); lanes 16-31 hold M=lane-16 (K=2,3). B (4x16, KxN) symmetric:
// lane carries N=lane&15, half-wave selects the K pair.
// C/D (16x16 f32): VGPR v -> row M = v + 8*(lane>=16), col N = lane&15.
__launch_bounds__(256)
__global__ void rgcn_gemm_scores(const float* __restrict__ x,
                                 const float* __restrict__ w,   // (16, 256)
                                 float* __restrict__ S,         // (N, 16)
                                 int N) {
    const int gwave = (blockIdx.x * blockDim.x + threadIdx.x) >> 5;
    const int lane  = threadIdx.x & 31;
    const int row0  = gwave * 16;
    if (row0 >= N) return;                      // wave-uniform: EXEC stays full

    const int m    = lane & 15;                 // node row within tile / relation col
    const int koff = (lane >> 4) * 2;           // half-wave K pair: {0,1} or {2,3}

    const int arow = min(row0 + m, N - 1);      // clamp (N%16==0 here, but safe)
    const float* __restrict__ xrow = x + (size_t)arow * DFEAT + koff;
    const float* __restrict__ wrow = w + (size_t)m    * DFEAT + koff;

    v8f c = {};
    #pragma unroll 16
    for (int k = 0; k < DFEAT; k += 4) {
        v2f a = *(const v2f*)(xrow + k);        // global_load_b64
        v2f b = *(const v2f*)(wrow + k);        // global_load_b64 (L2-hot, 16KB)
        c = __builtin_amdgcn_wmma_f32_16x16x4_f32(
                /*neg_a=*/false, a, /*neg_b=*/false, b,
                /*c_mod=*/(short)0, c, /*reuse_a=*/false, /*reuse_b=*/false);
    }

    const int rbase = row0 + (lane >> 4) * 8;   // C/D layout: M = v + 8*(lane>=16)
    float* __restrict__ Srow = S + (size_t)rbase * NREL + m;
    if (row0 + 16 <= N) {
        // Wave-uniform fast path: full tile, no exec-mask juggling.
        // Each store: 32 lanes x 4B = two contiguous 64B rows -> fully coalesced.
        #pragma unroll
        for (int v = 0; v < 8; ++v)
            Srow[(size_t)v * NREL] = c[v];
    } else {
        #pragma unroll
        for (int v = 0; v < 8; ++v)
            if (rbase + v < N) Srow[(size_t)v * NREL] = c[v];
    }
}

// ---- kernel 3: edge scatter: 4B gather + 2 fp32 atomics per edge ------------
__launch_bounds__(256)
__global__ void rgcn_edge_scatter(const int* __restrict__ edge_index, // (2,E)
                                  const int* __restrict__ edge_type,  // (E,)
                                  const float* __restrict__ S,        // (N,16)
                                  float* __restrict__ agg,
                                  float* __restrict__ cnt,
                                  int E) {
    int e = blockIdx.x * blockDim.x + threadIdx.x;
    if (e >= E) return;
    int src = edge_index[e];
    int dst = edge_index[E + e];
    int r   = edge_type[e];
    float msg = S[(size_t)src * NREL + r];
    atomicAdd(&agg[dst], msg);                  // global_atomic_add_f32
    atomicAdd(&cnt[dst], 1.0f);
}

// ---- kernel 4: one wave per node: x@root (wave reduce), tanh, scale --------
__launch_bounds__(256)
__global__ void rgcn_finalize(const float* __restrict__ x,
                              const float* __restrict__ root,   // (256,)
                              const float* __restrict__ bias,   // (1,)
                              const float* __restrict__ agg,
                              const float* __restrict__ cnt,
                              float* __restrict__ x_out,        // (N,256)
                              float* __restrict__ score,        // (N,)
                              int N) {
    const int node = (blockIdx.x * blockDim.x + threadIdx.x) >> 5;
    const int lane = threadIdx.x & 31;
    if (node >= N) return;

    const float* __restrict__ xr = x + (size_t)node * DFEAT;
    float xv[8];
    float dotv = 0.0f;
    #pragma unroll
    for (int i = 0; i < 8; ++i) {               // coalesced: lane + 32*i
        xv[i] = xr[lane + 32 * i];
        dotv  = fmaf(xv[i], root[lane + 32 * i], dotv);
    }
    // wave32 butterfly reduction (ds_swizzle/permute path)
    #pragma unroll
    for (int off = 16; off > 0; off >>= 1)
        dotv += __shfl_xor(dotv, off, 32);

    float c    = cnt[node];
    float mean = agg[node] / fmaxf(c, 1.0f);
    float s    = tanhf(mean + dotv + bias[0]);

    if (lane == 0) score[node] = s;
    float* __restrict__ o = x_out + (size_t)node * DFEAT;
    #pragma unroll
    for (int i = 0; i < 8; ++i) o[lane + 32 * i] = xv[i] * s;
}

// ---------------------------------------------------------------------------
extern "C" void kernel_launch(void* const* d_in, const int* in_sizes, int n_in,
                              void* d_out, int out_size, void* d_ws, size_t ws_size,
                              hipStream_t stream) {
    const float* x     = (const float*)d_in[0];   // (N,256) f32
    const int*   ei    = (const int*)  d_in[1];   // (2,E)   i32
    const int*   et    = (const int*)  d_in[2];   // (E,)    i32
    const float* w     = (const float*)d_in[3];   // (16,256,1) f32
    const float* root  = (const float*)d_in[4];   // (256,1) f32
    const float* bias  = (const float*)d_in[5];   // (1,)    f32

    const int N = in_sizes[0] / DFEAT;
    const int E = in_sizes[2];

    // workspace layout: S (N*16 f32) | agg (N f32) | cnt (N f32)
    float* S   = (float*)d_ws;
    float* agg = S + (size_t)N * NREL;
    float* cnt = agg + N;

    float* x_out = (float*)d_out;                 // (N,256)
    float* score = x_out + (size_t)N * DFEAT;     // (N,)

    // 1) zero agg+cnt (contiguous 2N floats)
    {
        int n = 2 * N;
        rgcn_zero<<<(n + 255) / 256, 256, 0, stream>>>(agg, n);
    }
    // 2) WMMA GEMM: one wave per 16-row tile, 8 waves (256 thr) per block
    {
        int tiles  = (N + 15) / 16;               // 3125 for N=50000
        int blocks = (tiles + 7) / 8;
        rgcn_gemm_scores<<<blocks, 256, 0, stream>>>(x, w, S, N);
    }
    // 3) edge scatter
    rgcn_edge_scatter<<<(E + 255) / 256, 256, 0, stream>>>(ei, et, S, agg, cnt, E);
    // 4) finalize: one wave per node
    {
        int blocks = (N + 7) / 8;                 // 8 waves of 32 per block
        rgcn_finalize<<<blocks, 256, 0, stream>>>(x, root, bias, agg, cnt,
                                                  x_out, score, N);
    }
}